// DualMem_54726473285695
// MI455X (gfx1250) — compile-verified
//
#include <hip/hip_runtime.h>
#include <math.h>

// ---------------------------------------------------------------------------
// DualMem fused kernel for MI455X (gfx1250, wave32)
//
// Roofline: dominant traffic = image_feature_memory (1000 x 50 x 1024 f32
// = 205 MB). 23.3 TB/s => ~9us floor if read exactly once. One workgroup per
// class; the whole 51x1024 f32 class tile is staged in LDS (320KB/WGP on
// CDNA5), loaded by the Tensor Data Mover (tensor_load_to_lds, iterate mode
// handles the bank-conflict row padding 1024->1028), and every phase is fused.
//
// Per-row math restructured into 5 streaming reductions over the raw mem row:
// mem.qn, mem.gk, mem.gv, sum(mem) [empty test], sum(mem^2), using
//   |K|^2 = sum(mem^2) + 2*mem.gk + |gk|^2   (gk terms are per-class scalars)
// The 4 "row dot fixed-vector" reductions run on the matrix pipe with
// V_WMMA_F32_16X16X4_F32 (A = 16 rows x 4 k, B cols = {qn, gk, gv, 1}),
// 128 chained WMMAs per wave; sum(mem^2) rides along on VALU (XDL WMMA
// co-executes with VALU). Empty rows have Vn==0 in the reference, so their
// weight is forced to 0 and they drop out.
// ---------------------------------------------------------------------------

typedef __attribute__((ext_vector_type(2))) float v2f;
typedef __attribute__((ext_vector_type(8))) float v8f;
typedef __attribute__((ext_vector_type(4))) unsigned int v4u;
typedef __attribute__((ext_vector_type(4))) int v4i;
typedef __attribute__((ext_vector_type(8))) int v8i;

#define N_CLASSES 1000
#define M_ROWS    51          // 50 memory slots + 1 fixed global row
#define D_DIM     1024
#define BETA_F    5.5f
#define STRIDEF   1028        // padded LDS row stride (floats) -> no bank conflicts
#define TILE_ROWS 64          // rows 51..63 zero-padded for the 4th WMMA tile

// ---- LDS carve (floats) ----------------------------------------------------
#define LDS_FLOATS (TILE_ROWS*STRIDEF + 3*D_DIM + 512 + 256 + 64 + 64)

// LDS byte offset of a shared-memory pointer (generic -> addrspace(3) -> int).
typedef __attribute__((address_space(3))) void lds_void_t;
__device__ __forceinline__ unsigned ldsOffsetOf(const void* p) {
  return (unsigned)(size_t)(lds_void_t*)(void*)(const_cast<void*>(p));
}

__device__ __forceinline__ float warpSum(float v) {
  #pragma unroll
  for (int o = 16; o; o >>= 1) v += __shfl_xor(v, o, 32);
  return v;
}
__device__ __forceinline__ float warpMax(float v) {
  #pragma unroll
  for (int o = 16; o; o >>= 1) v = fmaxf(v, __shfl_xor(v, o, 32));
  return v;
}
// Full-block reduction; every thread receives the total. Uniform control flow.
__device__ __forceinline__ float blockSum(float v, float* scr) {
  v = warpSum(v);
  const int w = threadIdx.x >> 5, l = threadIdx.x & 31, nw = blockDim.x >> 5;
  if (l == 0) scr[w] = v;
  __syncthreads();
  v = (l < nw) ? scr[l] : 0.0f;
  v = warpSum(v);
  __syncthreads();
  return v;
}
__device__ __forceinline__ float blockMax(float v, float* scr) {
  v = warpMax(v);
  const int w = threadIdx.x >> 5, l = threadIdx.x & 31, nw = blockDim.x >> 5;
  if (l == 0) scr[w] = v;
  __syncthreads();
  v = (l < nw) ? scr[l] : -__builtin_inff();
  v = warpMax(v);
  __syncthreads();
  return v;
}

// ---------------------------------------------------------------------------
// Kernel 1: qn = l2norm(img_feat + mean(global_bias, axis=0))   (1024 floats)
// ---------------------------------------------------------------------------
__global__ __launch_bounds__(1024) void prep_qn(const float* __restrict__ img,
                                                const float* __restrict__ gb,
                                                float* __restrict__ qn_out) {
  __shared__ float scr[32];
  const int d = threadIdx.x;
  float s = 0.0f;
  for (int c = 0; c < N_CLASSES; ++c) s += gb[c * D_DIM + d];
  const float q = img[d] + s * (1.0f / (float)N_CLASSES);
  const float nn = blockSum(q * q, scr);
  qn_out[d] = q * rsqrtf(nn);
}

// ---------------------------------------------------------------------------
// Kernel 2: per-class fused pipeline. grid = 1000 blocks, 256 threads (8 waves)
// ---------------------------------------------------------------------------
__global__ __launch_bounds__(256) void dualmem_main(
    const float* __restrict__ img,     // (D)
    const float* __restrict__ memf,    // (C,50,D)
    const float* __restrict__ fixedg,  // (C,1,D)
    const float* __restrict__ gk,      // (C,D)
    const float* __restrict__ gv,      // (C,D)
    const float* __restrict__ ffn,     // (C,D)
    const float* __restrict__ qn,      // (D)
    float* __restrict__ logits)        // (C) unscaled
{
  extern __shared__ float lds[];
  float* tile    = lds;                           // 64 x 1028
  float* qn_s    = tile + TILE_ROWS * STRIDEF;    // 1024
  float* gk_s    = qn_s + D_DIM;                  // 1024
  float* gv_s    = gk_s + D_DIM;                  // 1024
  float* dotred  = gv_s + D_DIM;                  // 512
  float* sqred   = dotred + 512;                  // 256
  float* w_s     = sqred + 256;                   // 64
  float* scratch = w_s + 64;                      // 64

  const int c    = blockIdx.x;
  const int tid  = threadIdx.x;
  const int lane = tid & 31;
  const int wave = tid >> 5;

  // ---- TDM: DMA the 50x1024 f32 block into LDS rows 0..49 (iterate mode:
  //      one 1024-elem row per iteration, LDS advances by the padded 1028). --
  if (tid < 32) {   // one issuing wave (TDM ignores EXEC; 1 op per wave)
    const unsigned long long ga =
        (unsigned long long)(size_t)(memf + (size_t)c * 50 * D_DIM);
    const unsigned ldsA = ldsOffsetOf(tile);
    const v4u g0 = { 1u,                                   // count=1 (valid D#)
                     ldsA,                                 // lds_addr (bytes)
                     (unsigned)ga,                         // global_addr[31:0]
                     (unsigned)(ga >> 32) | 0x80000000u }; // [56:32] | type=2
    const v8i g1 = { (int)((2u << 16) | (1u << 19)),       // data_size=4B, iterate
                     (int)(1024u << 16),                   // tensor_dim0 = 1024
                     (int)(1u << 16),                      // tensor_dim1 = 1
                     (int)(1024u << 16),                   // tile_dim0 = 1024
                     0,                                    // tile_dim1/2 unused
                     1024,                                 // tensor_dim0_stride
                     0, 0 };
    const v4i g2 = { 0,                                    // tensor_dim2
                     1028,                                 // lds_addr_increment
                     1024,                                 // global_addr_increment
                     (int)(49u << 16) };                   // iterate_count: 50 rows
    const v4i g3 = { 0, 0, 0, 0 };
    const v8i gx = { 0, 0, 0, 0, 0, 0, 0, 0 };             // 6-arg form: unused tail
    __builtin_amdgcn_tensor_load_to_lds(g0, g1, g2, g3, gx, 0);
  }

  // ---- stage qn/gk/gv into LDS; per-class scalar partials -----------------
  const float* gkc = gk + (size_t)c * D_DIM;
  const float* gvc = gv + (size_t)c * D_DIM;
  float p_gkq, p_gk2, p_gv2;
  {
    const int i = tid * 4;                         // 256 threads x 4 = 1024
    const float4 q4 = *(const float4*)(qn  + i);
    const float4 k4 = *(const float4*)(gkc + i);
    const float4 v4 = *(const float4*)(gvc + i);
    *(float4*)(qn_s + i) = q4;
    *(float4*)(gk_s + i) = k4;
    *(float4*)(gv_s + i) = v4;
    p_gkq = q4.x*k4.x + q4.y*k4.y + q4.z*k4.z + q4.w*k4.w;
    p_gk2 = k4.x*k4.x + k4.y*k4.y + k4.z*k4.z + k4.w*k4.w;
    p_gv2 = v4.x*v4.x + v4.y*v4.y + v4.z*v4.z + v4.w*v4.w;
  }

  // ---- row 50 (fixed global feat) + zero pad rows 51..63 ------------------
  {
    const float* fixc = fixedg + (size_t)c * D_DIM;
    const int i = tid * 4;
    *(float4*)(tile + 50 * STRIDEF + i) = *(const float4*)(fixc + i);
    for (int f = tid; f < ((TILE_ROWS - M_ROWS) * D_DIM) / 4; f += 256) {
      const int e = f * 4, row = M_ROWS + (e >> 10), col = e & (D_DIM - 1);
      *(float4*)(tile + row * STRIDEF + col) = make_float4(0.f, 0.f, 0.f, 0.f);
    }
  }

  const float cgkq = blockSum(p_gkq, scratch);   // qn . gk
  const float cgk2 = blockSum(p_gk2, scratch);   // |gk|^2
  const float cgv2 = blockSum(p_gv2, scratch);   // |gv|^2

  if (tid < 32) __builtin_amdgcn_s_wait_tensorcnt((short)0);  // DMA done
  __syncthreads();                               // tile + vectors visible

  // ---- Phase A: WMMA row reductions ---------------------------------------
  // wave -> (m-tile 0..3, k-half 0..1); 128 chained V_WMMA_F32_16X16X4_F32.
  // Uniform trip count; lane-dependent start folded into the pointers.
  const int tileIdx = wave & 3;
  const int half    = wave >> 2;
  const int n       = lane & 15;                         // D-matrix column
  const int rowM    = tileIdx * 16 + (lane & 15);        // A-matrix row
  const int k0base  = (lane >> 4) << 1;                  // 0 or 2 (K lane-half)
  // B columns: 0->qn, 1->gk, 2->gv, 3->ones, 4..15 -> zero
  const float* bsel  = (n == 1) ? gk_s : ((n == 2) ? gv_s : qn_s);
  const float  bmask = (n < 3) ? 1.0f : 0.0f;
  const float  badd  = (n == 3) ? 1.0f : 0.0f;
  const int    kofs  = half * 512 + k0base;
  const float* ap    = tile + rowM * STRIDEF + kofs;
  const float* bp    = bsel + kofs;

  v8f   acc = {};
  float sq  = 0.0f;
  #pragma unroll 4
  for (int ch = 0; ch < 128; ++ch) {
    const v2f a = *(const v2f*)ap;                       // ds_load_b64
    v2f b;
    b.x = bp[0] * bmask + badd;
    b.y = bp[1] * bmask + badd;
    sq  = fmaf(a.x, a.x, sq);
    sq  = fmaf(a.y, a.y, sq);
    acc = __builtin_amdgcn_wmma_f32_16x16x4_f32(
        /*neg_a=*/false, a, /*neg_b=*/false, b,
        /*c_mod=*/(short)0, acc, /*reuse_a=*/false, /*reuse_b=*/false);
    ap += 4; bp += 4;
  }

  // scatter partials: D[r][n] lanes0-15 -> M=r, lanes16-31 -> M=r+8
  {
    const int rbase = (lane >= 16) ? 8 : 0;
    if (n < 4) {
      #pragma unroll
      for (int r = 0; r < 8; ++r)
        dotred[(((tileIdx * 2 + half) * 16) + r + rbase) * 4 + n] = acc[r];
    }
    sqred[(tileIdx * 2 + half) * 32 + lane] = sq;
  }
  __syncthreads();

  // ---- weights per row -----------------------------------------------------
  if (tid < M_ROWS) {
    const int m = tid, t4 = m >> 4, rl = m & 15;
    const float* d0 = &dotred[((t4 * 2 + 0) * 16 + rl) * 4];
    const float* d1 = &dotred[((t4 * 2 + 1) * 16 + rl) * 4];
    const float dotQ   = d0[0] + d1[0];
    const float dotGK  = d0[1] + d1[1];
    const float dotGV  = d0[2] + d1[2];
    const float rowsum = d0[3] + d1[3];
    const float sumsq  = sqred[(t4 * 2 + 0) * 32 + rl] + sqred[(t4 * 2 + 0) * 32 + rl + 16]
                       + sqred[(t4 * 2 + 1) * 32 + rl] + sqred[(t4 * 2 + 1) * 32 + rl + 16];
    float w = 0.0f;
    if (rowsum != 0.0f) {                       // empty slot -> Vn==0 -> drops
      const float kq  = dotQ + cgkq;
      const float k2  = sumsq + 2.0f * dotGK + cgk2;
      const float v2  = sumsq + 2.0f * dotGV + cgv2;
      const float sim = __expf(-BETA_F * (1.0f - kq * rsqrtf(k2)));
      w = sim * rsqrtf(v2);                     // fold 1/|V| into the weight
    }
    w_s[m] = w;
  }
  __syncthreads();

  // ---- Phase B: adaptive[d] = sum_m w[m]*mem[m,d] + (sum_m w[m])*gv[d] ----
  const int d0i = tid * 4;
  float a0 = 0.f, a1 = 0.f, a2 = 0.f, a3 = 0.f, wsum = 0.f;
  #pragma unroll 3
  for (int m = 0; m < M_ROWS; ++m) {
    const float  wm = w_s[m];
    const float4 r4 = *(const float4*)(tile + m * STRIDEF + d0i);
    a0 = fmaf(wm, r4.x, a0);
    a1 = fmaf(wm, r4.y, a1);
    a2 = fmaf(wm, r4.z, a2);
    a3 = fmaf(wm, r4.w, a3);
    wsum += wm;
  }
  {
    const float4 g4 = *(const float4*)(gv_s + d0i);
    a0 += wsum * g4.x;  a1 += wsum * g4.y;
    a2 += wsum * g4.z;  a3 += wsum * g4.w;
  }

  // ---- Phase C: fused double-normalization + logit ------------------------
  // t = a/|a| + f ;  |t|^2 = 1 + 2*(a.f)/|a| + |f|^2 ; t.i = (a.i)/|a| + f.i
  const float4 f4 = *(const float4*)(ffn + (size_t)c * D_DIM + d0i);
  const float4 i4 = *(const float4*)(img + d0i);
  float s_aa = a0*a0 + a1*a1 + a2*a2 + a3*a3;
  float s_af = a0*f4.x + a1*f4.y + a2*f4.z + a3*f4.w;
  float s_ai = a0*i4.x + a1*i4.y + a2*i4.z + a3*i4.w;
  float s_ff = f4.x*f4.x + f4.y*f4.y + f4.z*f4.z + f4.w*f4.w;
  float s_fi = f4.x*i4.x + f4.y*i4.y + f4.z*i4.z + f4.w*i4.w;
  s_aa = blockSum(s_aa, scratch);
  s_af = blockSum(s_af, scratch);
  s_ai = blockSum(s_ai, scratch);
  s_ff = blockSum(s_ff, scratch);
  s_fi = blockSum(s_fi, scratch);
  if (tid == 0) {
    const float inv1 = rsqrtf(s_aa);
    const float t2   = 1.0f + 2.0f * s_af * inv1 + s_ff;
    const float tdot = s_ai * inv1 + s_fi;
    logits[c] = tdot * rsqrtf(t2);
  }
}

// ---------------------------------------------------------------------------
// Kernel 3: softmax over 1000 scaled logits
// ---------------------------------------------------------------------------
__global__ __launch_bounds__(1024) void softmax_k(const float* __restrict__ lg,
                                                  const float* __restrict__ logit_scale,
                                                  float* __restrict__ out) {
  __shared__ float scr[32];
  const int t = threadIdx.x;
  const float ls = __expf(logit_scale[0]);
  const float v  = (t < N_CLASSES) ? lg[t] * ls : -__builtin_inff();
  const float mx = blockMax(v, scr);
  const float e  = (t < N_CLASSES) ? __expf(v - mx) : 0.0f;
  const float se = blockSum(e, scr);
  if (t < N_CLASSES) out[t] = e / se;
}

// ---------------------------------------------------------------------------
extern "C" void kernel_launch(void* const* d_in, const int* in_sizes, int n_in,
                              void* d_out, int out_size, void* d_ws, size_t ws_size,
                              hipStream_t stream) {
  const float* img    = (const float*)d_in[0];   // (1,1024)
  const float* memf   = (const float*)d_in[1];   // (1000,50,1024)
  const float* fixedg = (const float*)d_in[2];   // (1000,1,1024)
  const float* gb     = (const float*)d_in[3];   // (1000,1024)
  const float* gk     = (const float*)d_in[4];   // (1000,1024)
  const float* gv     = (const float*)d_in[5];   // (1000,1024)
  const float* ffn    = (const float*)d_in[6];   // (1000,1024)
  const float* ls     = (const float*)d_in[7];   // scalar
  float* out = (float*)d_out;                    // (1,1000) f32

  float* qn_ws     = (float*)d_ws;               // 1024 floats
  float* logits_ws = qn_ws + D_DIM;              // 1000 floats

  const size_t ldsBytes = (size_t)LDS_FLOATS * sizeof(float);  // ~279 KB < 320 KB
  (void)hipFuncSetAttribute((const void*)dualmem_main,
                            hipFuncAttributeMaxDynamicSharedMemorySize,
                            (int)ldsBytes);

  prep_qn<<<1, 1024, 0, stream>>>(img, gb, qn_ws);
  dualmem_main<<<N_CLASSES, 256, ldsBytes, stream>>>(img, memf, fixedg, gk, gv,
                                                     ffn, qn_ws, logits_ws);
  softmax_k<<<1, 1024, 0, stream>>>(logits_ws, ls, out);
}